// ChamferLoss_8194797601432
// MI455X (gfx1250) — compile-verified
//
#include <hip/hip_runtime.h>

typedef __attribute__((ext_vector_type(2))) float v2f;
typedef __attribute__((ext_vector_type(8))) float v8f;

#define CHUNK_PTS 2048
#define REF_PER_BLOCK 128
#define THREADS 256
#define WAVES 8

// D = A(16x4) x B(4x16) + C via v_wmma_f32_16x16x4_f32.
// A rows = src points (minimized axis), scaled (-2x,-2y,-2z,1).
// B cols = ref points (x,y,z,||ref||^2)  =>  acc = ||ref||^2 - 2 src.ref.
// Distance = acc + ||src||^2, folded with a single add+min per element.
__global__ __launch_bounds__(THREADS)
void chamfer_min_kernel(const float* __restrict__ pcA,   // output_pc, [N,3]
                        const float* __restrict__ pcB,   // gt_pc,     [N,3]
                        float* __restrict__ partial,     // one float per block
                        int nchunks, int ntilesJ)
{
    __shared__ float4 sA[CHUNK_PTS];   // (-2x,-2y,-2z,1) per src point  (32 KB)
    __shared__ float  sN[CHUNK_PTS];   // ||src||^2                      ( 8 KB)
    __shared__ float  sWave[WAVES];

    const int b     = blockIdx.x;
    const int per   = nchunks * ntilesJ;
    const int dir   = b / per;
    const int rem   = b % per;
    const int chunk = rem / ntilesJ;
    const int jtile = rem % ntilesJ;

    const float* src = dir ? pcB : pcA;   // the chunked cloud
    const float* ref = dir ? pcA : pcB;

    // ---- stage src chunk into LDS (pre-scaled) ----
    const int tid = threadIdx.x;
    const float* sbase = src + (size_t)chunk * CHUNK_PTS * 3;
    for (int i = tid; i < CHUNK_PTS; i += THREADS) {
        float x = sbase[i * 3 + 0];
        float y = sbase[i * 3 + 1];
        float z = sbase[i * 3 + 2];
        sA[i] = make_float4(-2.f * x, -2.f * y, -2.f * z, 1.f);
        sN[i] = x * x + y * y + z * z;
    }
    __syncthreads();

    const int wave = tid >> 5;
    const int lane = tid & 31;
    const int lo   = lane & 15;
    const int hi   = lane >> 4;

    // ---- B fragment: 16 ref points owned by this wave (kept in VGPRs) ----
    const int j = jtile * REF_PER_BLOCK + wave * 16 + lo;
    const float rx = ref[j * 3 + 0];
    const float ry = ref[j * 3 + 1];
    const float rz = ref[j * 3 + 2];
    v2f bfrag;
    if (hi) { bfrag[0] = rz; bfrag[1] = rx * rx + ry * ry + rz * rz; }
    else    { bfrag[0] = rx; bfrag[1] = ry; }

    // Per-lane LDS base for the A fragment: point (lane&15), floats [0,1] or [2,3].
    const float* aLane = ((const float*)&sA[lo]) + hi * 2;

    float m = 3.0e38f;
#pragma unroll 4
    for (int t = 0; t < CHUNK_PTS / 16; ++t) {
        // A fragment: ds_load_b64 per lane (tile stride = 16 pts * 16B = 64 floats)
        const float2 av = *(const float2*)(aLane + t * 64);
        v2f afrag;
        afrag[0] = av.x;
        afrag[1] = av.y;

        v8f c = {0.f, 0.f, 0.f, 0.f, 0.f, 0.f, 0.f, 0.f};
        c = __builtin_amdgcn_wmma_f32_16x16x4_f32(
                /*neg_a=*/false, afrag, /*neg_b=*/false, bfrag,
                /*c_mod=*/(short)0, c, /*reuse_a=*/false, /*reuse_b=*/false);

        // 8 contiguous src norms for rows M = t*16 + hi*8 + {0..7}
        const float4* np_ = (const float4*)(&sN[t * 16 + hi * 8]);
        const float4 s0 = np_[0];
        const float4 s1 = np_[1];
        m = fminf(m, c[0] + s0.x);
        m = fminf(m, c[1] + s0.y);
        m = fminf(m, c[2] + s0.z);
        m = fminf(m, c[3] + s0.w);
        m = fminf(m, c[4] + s1.x);
        m = fminf(m, c[5] + s1.y);
        m = fminf(m, c[6] + s1.z);
        m = fminf(m, c[7] + s1.w);
    }

    // lanes l and l^16 hold the same ref column (row halves): merge mins
    m = fminf(m, __shfl_xor(m, 16, 32));

    // sum the 16 per-ref minima (lanes 0..15 reduce among themselves)
    float s = m;
    s += __shfl_xor(s, 1, 32);
    s += __shfl_xor(s, 2, 32);
    s += __shfl_xor(s, 4, 32);
    s += __shfl_xor(s, 8, 32);
    if (lane == 0) sWave[wave] = s;
    __syncthreads();

    if (tid == 0) {
        float acc = 0.f;
        for (int w = 0; w < WAVES; ++w) acc += sWave[w];
        partial[b] = acc;
    }
}

__global__ __launch_bounds__(256)
void chamfer_reduce_kernel(const float* __restrict__ partial,
                           float* __restrict__ out,
                           int nparts, float inv)
{
    __shared__ float sbuf[256];
    const int tid = threadIdx.x;
    float s = 0.f;
    for (int i = tid; i < nparts; i += 256) s += partial[i];  // fixed order
    sbuf[tid] = s;
    __syncthreads();
    for (int off = 128; off > 0; off >>= 1) {
        if (tid < off) sbuf[tid] += sbuf[tid + off];
        __syncthreads();
    }
    if (tid == 0) out[0] = sbuf[0] * inv;
}

extern "C" void kernel_launch(void* const* d_in, const int* in_sizes, int n_in,
                              void* d_out, int out_size, void* d_ws, size_t ws_size,
                              hipStream_t stream) {
    const float* pcA = (const float*)d_in[0];   // output_pc
    const float* pcB = (const float*)d_in[1];   // gt_pc
    float* out = (float*)d_out;
    float* partial = (float*)d_ws;

    const int npts    = in_sizes[0] / 3;            // 16384
    const int nchunks = npts / CHUNK_PTS;           // 8
    const int ntilesJ = npts / REF_PER_BLOCK;       // 128
    const int nblocks = 2 * nchunks * ntilesJ;      // 2048

    chamfer_min_kernel<<<nblocks, THREADS, 0, stream>>>(pcA, pcB, partial,
                                                        nchunks, ntilesJ);

    const float inv = 1.0f / ((float)nchunks * (float)npts);  // mean1+mean2 share denom
    chamfer_reduce_kernel<<<1, 256, 0, stream>>>(partial, out, nblocks, inv);
}